// qrnn7_90494960927123
// MI455X (gfx1250) — compile-verified
//
#include <hip/hip_runtime.h>
#include <hip/hip_bf16.h>
#include <math.h>

typedef __attribute__((ext_vector_type(2))) float v2f;
typedef __attribute__((ext_vector_type(8))) float v8f;

constexpr int B_   = 128;
constexpr int T_   = 512;
constexpr int E_   = 12;
constexpr int H_   = 512;
constexpr int DACT = 10;
constexpr int OUTD = 64;
constexpr int G3H  = 3 * H_;   // 1536
constexpr int DH   = 2 * H_;   // 1024

// Branchless fast activations on the native TRANS units (v_exp_f32/v_rcp_f32).
// sigmoid: 1/(1+e^-x).  tanh: 1 - 2/(1+e^{2x}) -- exact limits at +-inf.
__device__ __forceinline__ float fast_sigmoid(float x) {
  return __builtin_amdgcn_rcpf(1.0f + __expf(-x));
}
__device__ __forceinline__ float fast_tanh(float x) {
  return 1.0f - 2.0f * __builtin_amdgcn_rcpf(1.0f + __expf(2.0f * x));
}

// ---------------------------------------------------------------------------
// Kernel 1: embedding gather + time-feature concat.
// enc layout: (T, B, E) fp32 -> enc[(t*B + b)*E + k]
// ---------------------------------------------------------------------------
__global__ void k_encode(const float* __restrict__ x, const float* __restrict__ emb,
                         float* __restrict__ enc) {
  int i = blockIdx.x * blockDim.x + threadIdx.x;   // i = b*T + t
  if (i >= B_ * T_) return;
  int b = i / T_, t = i % T_;
  const float* xp = x + (size_t)i * 3;
  float f0 = xp[0], f1 = xp[1];
  int idx = (int)xp[2];
  float* ep = enc + ((size_t)t * B_ + b) * E_;
  ep[0] = f0;
  ep[1] = f1;
  const float* er = emb + (size_t)idx * DACT;
#pragma unroll
  for (int k = 0; k < DACT; ++k) ep[2 + k] = er[k];
}

// ---------------------------------------------------------------------------
// Kernel 2: fused QRNN direction (GEMM via fp32 WMMA + branchless gates +
// fo-pool recurrence in registers + ReLU + store).
// Each wave owns a (16-batch x 16-hidden) tile; block = 4 waves (64 h-cols).
// grid = (H/64, B/16, 2 directions).
// out layout: (B, T, 2H) fp32; fwd -> cols [0,H), bwd -> cols [H,2H).
// ---------------------------------------------------------------------------
__global__ void __launch_bounds__(128) k_qrnn(const float* __restrict__ enc,
                                              const float* __restrict__ Wf,
                                              const float* __restrict__ bf,
                                              const float* __restrict__ Wb,
                                              const float* __restrict__ bb,
                                              float* __restrict__ out) {
  const int lane = threadIdx.x & 31;
  const int wave = threadIdx.x >> 5;
  const int dir  = blockIdx.z;
  const int h0   = (blockIdx.x * 4 + wave) * 16;
  const int b0   = blockIdx.y * 16;
  const int nn   = lane & 15;       // column within tile (N) / A row (M)
  const int half = lane >> 4;       // half-wave id
  const int krow = half * 2;        // K offset for this half (16x4 f32 A/B layout)

  const float* W    = dir ? Wb : Wf;
  const float* bias = dir ? bb : bf;

  // Preload B-matrix tiles (time-invariant): 3 gates x 3 K-chunks of 4.
  // B 4x16 f32 layout: VGPR j, lane L -> K = j + 2*(L>>4), N = L&15.
  v2f Bv[3][3];
#pragma unroll
  for (int g = 0; g < 3; ++g) {
    int col = g * H_ + h0 + nn;
#pragma unroll
    for (int kc = 0; kc < 3; ++kc) {
      int kb = kc * 4 + krow;
      Bv[g][kc].x = W[(size_t)kb * G3H + col];
      Bv[g][kc].y = W[(size_t)(kb + 1) * G3H + col];
    }
  }

  // Bias-seeded accumulator templates (same value down each column).
  v8f Cz0, Cf0, Co0;
  {
    float bz = bias[0 * H_ + h0 + nn];
    float bg = bias[1 * H_ + h0 + nn];
    float bo = bias[2 * H_ + h0 + nn];
#pragma unroll
    for (int r = 0; r < 8; ++r) { Cz0[r] = bz; Cf0[r] = bg; Co0[r] = bo; }
  }

  // fo-pool cell state lives in registers across the whole T loop.
  v8f c;
#pragma unroll
  for (int r = 0; r < 8; ++r) c[r] = 0.0f;

  // Per-lane output base offsets: C/D element r -> batch m = r + 8*half.
  size_t obase[8];
  {
    int dcol = dir * H_ + h0 + nn;
#pragma unroll
    for (int r = 0; r < 8; ++r) {
      int m = r + half * 8;
      obase[r] = (size_t)(b0 + m) * T_ * DH + dcol;
    }
  }

  // A-row pointer for this lane (batch b0+nn, K offset krow).
  const float* encA = enc + ((size_t)(b0 + nn)) * E_ + krow;
  const size_t stepT = (size_t)B_ * E_;

  for (int s = 0; s < T_; ++s) {
    int t = dir ? (T_ - 1 - s) : s;
    const float* pa = encA + (size_t)t * stepT;

    // Prefetch next timestep's A row (global_prefetch_b8).
    int tn = dir ? (t > 0 ? t - 1 : t) : (t + 1 < T_ ? t + 1 : t);
    __builtin_prefetch(encA + (size_t)tn * stepT, 0, 1);

    v2f Av[3];
#pragma unroll
    for (int kc = 0; kc < 3; ++kc) {
      Av[kc].x = pa[kc * 4];
      Av[kc].y = pa[kc * 4 + 1];
    }

    // Y = enc_t @ W + b for the three gates: 3 chained K=4 fp32 WMMAs each.
    v8f yz = Cz0, yf = Cf0, yo = Co0;
#pragma unroll
    for (int kc = 0; kc < 3; ++kc)
      yz = __builtin_amdgcn_wmma_f32_16x16x4_f32(false, Av[kc], false, Bv[0][kc],
                                                 (short)0, yz, false, false);
#pragma unroll
    for (int kc = 0; kc < 3; ++kc)
      yf = __builtin_amdgcn_wmma_f32_16x16x4_f32(false, Av[kc], false, Bv[1][kc],
                                                 (short)0, yf, false, false);
#pragma unroll
    for (int kc = 0; kc < 3; ++kc)
      yo = __builtin_amdgcn_wmma_f32_16x16x4_f32(false, Av[kc], false, Bv[2][kc],
                                                 (short)0, yo, false, false);

    // Branchless gates + recurrence + ReLU, lane-local on the C/D layout.
    size_t trow = (size_t)t * DH;
#pragma unroll
    for (int r = 0; r < 8; ++r) {
      float z  = fast_tanh(yz[r]);
      float fg = fast_sigmoid(yf[r]);
      float og = fast_sigmoid(yo[r]);
      float cv = fmaf(fg, c[r] - z, z);   // f*c + (1-f)*z
      c[r] = cv;
      float h = og * cv;
      out[obase[r] + trow] = fmaxf(h, 0.0f);
    }
  }
}

// ---------------------------------------------------------------------------
// Kernel 3: per-batch attention softmax over T + weighted sum + ReLU + GEMV.
// One block (256 threads) per batch row; streams `out` twice (bandwidth bound).
// ---------------------------------------------------------------------------
__global__ void __launch_bounds__(256) k_attn(const float* __restrict__ out,
                                              const float* __restrict__ Mu,
                                              const float* __restrict__ Wout,
                                              const float* __restrict__ bout,
                                              float* __restrict__ res) {
  __shared__ float s_attn[T_];
  __shared__ float s_red[256];
  __shared__ float s_ctx[DH];

  const int tid = threadIdx.x;
  const int b   = blockIdx.x;
  const float* ob = out + (size_t)b * T_ * DH;

  // Pass 1: logits[t] = out[b,t,:] . Mu
  for (int t = tid; t < T_; t += 256) {
    const float* row = ob + (size_t)t * DH;
    float acc = 0.0f;
    for (int d = 0; d < DH; d += 4) {
      float4 v = *(const float4*)(row + d);
      float4 m = *(const float4*)(Mu + d);
      acc += v.x * m.x + v.y * m.y + v.z * m.z + v.w * m.w;
    }
    s_attn[t] = acc;
  }
  __syncthreads();

  // Softmax over T: max-reduce
  float lmax = -3.402823466e38f;
  for (int t = tid; t < T_; t += 256) lmax = fmaxf(lmax, s_attn[t]);
  s_red[tid] = lmax;
  __syncthreads();
  for (int st = 128; st > 0; st >>= 1) {
    if (tid < st) s_red[tid] = fmaxf(s_red[tid], s_red[tid + st]);
    __syncthreads();
  }
  float gmax = s_red[0];
  __syncthreads();

  // exp + sum-reduce
  float lsum = 0.0f;
  for (int t = tid; t < T_; t += 256) {
    float e = __expf(s_attn[t] - gmax);
    s_attn[t] = e;
    lsum += e;
  }
  s_red[tid] = lsum;
  __syncthreads();
  for (int st = 128; st > 0; st >>= 1) {
    if (tid < st) s_red[tid] += s_red[tid + st];
    __syncthreads();
  }
  float inv = 1.0f / s_red[0];
  __syncthreads();
  for (int t = tid; t < T_; t += 256) s_attn[t] *= inv;
  __syncthreads();

  // Pass 2: context[d] = relu(sum_t attn[t] * out[b,t,d]); 4 dims per thread.
  int d0 = tid * 4;
  float4 acc = make_float4(0.f, 0.f, 0.f, 0.f);
  for (int t = 0; t < T_; ++t) {
    float a = s_attn[t];
    float4 v = *(const float4*)(ob + (size_t)t * DH + d0);
    acc.x += a * v.x; acc.y += a * v.y; acc.z += a * v.z; acc.w += a * v.w;
  }
  s_ctx[d0 + 0] = fmaxf(acc.x, 0.f);
  s_ctx[d0 + 1] = fmaxf(acc.y, 0.f);
  s_ctx[d0 + 2] = fmaxf(acc.z, 0.f);
  s_ctx[d0 + 3] = fmaxf(acc.w, 0.f);
  __syncthreads();

  // Pass 3: result = context @ W_out^T + b_out (64 outputs)
  if (tid < OUTD) {
    const float* w = Wout + (size_t)tid * DH;
    float a2 = bout[tid];
    for (int d = 0; d < DH; ++d) a2 += s_ctx[d] * w[d];
    res[(size_t)b * OUTD + tid] = a2;
  }
}

// ---------------------------------------------------------------------------
extern "C" void kernel_launch(void* const* d_in, const int* in_sizes, int n_in,
                              void* d_out, int out_size, void* d_ws, size_t ws_size,
                              hipStream_t stream) {
  const float* x    = (const float*)d_in[0];
  const float* emb  = (const float*)d_in[1];
  const float* Wf   = (const float*)d_in[2];
  const float* bf   = (const float*)d_in[3];
  const float* Wb   = (const float*)d_in[4];
  const float* bb   = (const float*)d_in[5];
  const float* Mu   = (const float*)d_in[6];
  const float* Wout = (const float*)d_in[7];
  const float* bout = (const float*)d_in[8];
  float* res = (float*)d_out;

  float* ws  = (float*)d_ws;
  float* enc = ws;                                   // T*B*E  floats (~3 MB)
  float* out = ws + (size_t)T_ * B_ * E_;            // B*T*2H floats (~268 MB)

  k_encode<<<(B_ * T_ + 255) / 256, 256, 0, stream>>>(x, emb, enc);
  k_qrnn<<<dim3(H_ / 64, B_ / 16, 2), 128, 0, stream>>>(enc, Wf, bf, Wb, bb, out);
  k_attn<<<B_, 256, 0, stream>>>(out, Mu, Wout, bout, res);
}